// LoRALinear_3530463117951
// MI455X (gfx1250) — compile-verified
//
#include <hip/hip_runtime.h>

#define IN_F   1024
#define OUT_F  1024
#define RANK   8
#define BATCH  8192

typedef float v2f __attribute__((ext_vector_type(2)));
typedef float v4f __attribute__((ext_vector_type(4)));
typedef float v8f __attribute__((ext_vector_type(8)));

// ---------------------------------------------------------------------------
// Kernel 1: t[b, r] = sum_i x[b, i] * A[dom[b]][i*RANK + r]
// One wave (32 lanes) per sample row; coalesced b128 reads of LoRA-A rows,
// wave32 xor-shuffle reduction.
// ---------------------------------------------------------------------------
__global__ __launch_bounds__(256) void lora_t_kernel(
    const float* __restrict__ x, const int* __restrict__ dom,
    const float* __restrict__ Aw, float* __restrict__ t) {
  const int wave = threadIdx.x >> 5;
  const int lane = threadIdx.x & 31;
  const int row  = (blockIdx.x << 3) + wave;

  const int d = dom[row];
  const float* __restrict__ A  = Aw + (size_t)d * (IN_F * RANK);
  const float* __restrict__ xr = x + (size_t)row * IN_F;

  float acc[RANK];
#pragma unroll
  for (int r = 0; r < RANK; ++r) acc[r] = 0.0f;

#pragma unroll 4
  for (int c = 0; c < IN_F / 32; ++c) {
    const int i = (c << 5) + lane;
    const float xv = xr[i];
    const v4f a0 = *(const v4f*)(A + (size_t)i * RANK);
    const v4f a1 = *(const v4f*)(A + (size_t)i * RANK + 4);
#pragma unroll
    for (int r = 0; r < 4; ++r) {
      acc[r]     += xv * a0[r];
      acc[r + 4] += xv * a1[r];
    }
  }

#pragma unroll
  for (int off = 16; off > 0; off >>= 1) {
#pragma unroll
    for (int r = 0; r < RANK; ++r)
      acc[r] += __shfl_xor(acc[r], off, 32);
  }
  if (lane < RANK) t[(size_t)row * RANK + lane] = acc[lane];
}

// ---------------------------------------------------------------------------
// Kernel 2: out = x @ W^T + bias + (t * B[dom]) fused epilogue.
// 128x128 tile / WG, 8 wave32 waves, f32 WMMA 16x16x4.
// Double-buffered LDS staged with GLOBAL_LOAD_ASYNC_TO_LDS_B128 (ASYNCcnt),
// no VGPR round-trip for tile data.
// ---------------------------------------------------------------------------
#define BM 128
#define BN 128
#define BK 32
#define LDS_STRIDE 36            // +4 pad: 144B rows keep b128 dests 16B-aligned,
                                 // 36*m mod 64 distinct for m=0..15 (no conflicts)
#define NCHUNK   (IN_F / BK)     // 32
#define TILEELTS ((BM + BN) * LDS_STRIDE)
#define BUFBYTES (TILEELTS * 4)  // 36864 bytes per double-buffer slot

__global__ __launch_bounds__(256) void lora_gemm_kernel(
    const float* __restrict__ x, const int* __restrict__ dom,
    const float* __restrict__ W, const float* __restrict__ bias,
    const float* __restrict__ Bw, const float* __restrict__ t,
    float* __restrict__ out) {
  __shared__ float smem[2 * TILEELTS];   // 72 KB

  const int tid  = threadIdx.x;
  const int wave = tid >> 5;
  const int lane = tid & 31;
  const int rowBase = blockIdx.y * BM;
  const int colBase = blockIdx.x * BN;

  // loader mapping: thread -> (row 0..127, 16-col half); 4x b128 per tile
  const int ldRow = tid >> 1;
  const int ldCol = (tid & 1) << 4;

  // byte offsets (32-bit: max 33.5 MB)
  const unsigned xRowByte = (unsigned)(((rowBase + ldRow) * IN_F + ldCol) * 4);
  const unsigned wRowByte = (unsigned)(((colBase + ldRow) * IN_F + ldCol) * 4);
  // LDS byte address = low 32 bits of generic pointer (ISA: LDS aperture lo dword)
  const unsigned smemBase = (unsigned)(size_t)(const void*)smem;
  const unsigned ldsRowOff = (unsigned)((ldRow * LDS_STRIDE + ldCol) * 4);
  const unsigned xLdsBase = smemBase + ldsRowOff;
  const unsigned wLdsBase = smemBase + (unsigned)(BM * LDS_STRIDE * 4) + ldsRowOff;

  v8f c[8];
#pragma unroll
  for (int nt = 0; nt < 8; ++nt) {
#pragma unroll
    for (int j = 0; j < 8; ++j) c[nt][j] = 0.0f;
  }

  // WMMA f32 16x16x4 fragment addressing (ISA 7.12.2)
  const int m    = lane & 15;
  const int half = lane >> 4;
  const int koff = half << 1;

  // issue 8 async b128 loads (4 x-tile + 4 w-tile) for one K-chunk
  auto stage = [&](int bufIdx, unsigned kByte) {
    const unsigned bOff = (unsigned)bufIdx * (unsigned)BUFBYTES;
    const unsigned xl = xLdsBase + bOff;
    const unsigned wl = wLdsBase + bOff;
    const unsigned xo = xRowByte + kByte;
    const unsigned wo = wRowByte + kByte;
#pragma unroll
    for (int j = 0; j < 4; ++j) {
      asm volatile("global_load_async_to_lds_b128 %0, %1, %2"
                   :: "v"(xl + 16u * j), "v"(xo + 16u * j), "s"(x) : "memory");
      asm volatile("global_load_async_to_lds_b128 %0, %1, %2"
                   :: "v"(wl + 16u * j), "v"(wo + 16u * j), "s"(W) : "memory");
    }
  };

  stage(0, 0u);  // prologue

  for (int chunk = 0; chunk < NCHUNK; ++chunk) {
    const int cur = chunk & 1;
    if (chunk + 1 < NCHUNK) {
      stage(cur ^ 1, (unsigned)((chunk + 1) * BK * 4));
      // async loads complete in order: <=8 outstanding => current tile landed
      asm volatile("s_wait_asynccnt 0x8" ::: "memory");
    } else {
      asm volatile("s_wait_asynccnt 0x0" ::: "memory");
    }
    __syncthreads();

    const float* xsb = smem + cur * TILEELTS;
    const float* wsb = xsb + BM * LDS_STRIDE;
#pragma unroll
    for (int kk = 0; kk < BK; kk += 4) {
      const v2f a = *(const v2f*)(xsb + (wave * 16 + m) * LDS_STRIDE + kk + koff);
#pragma unroll
      for (int nt = 0; nt < 8; ++nt) {
        const v2f bf = *(const v2f*)(wsb + (nt * 16 + m) * LDS_STRIDE + kk + koff);
        c[nt] = __builtin_amdgcn_wmma_f32_16x16x4_f32(
            false, a, false, bf, (short)0, c[nt], false, false);
      }
    }
    __syncthreads();
  }

  // Epilogue: bias + rank-8 LoRA correction (B row reads coalesced per 16-lane half)
#pragma unroll
  for (int v = 0; v < 8; ++v) {
    const int row = rowBase + wave * 16 + half * 8 + v;
    const int d   = dom[row];
    const float* __restrict__ Bm = Bw + (size_t)d * (RANK * OUT_F);
    float tv[RANK];
#pragma unroll
    for (int r = 0; r < RANK; ++r) tv[r] = t[(size_t)row * RANK + r];
#pragma unroll
    for (int nt = 0; nt < 8; ++nt) {
      const int col = colBase + nt * 16 + m;
      float acc = c[nt][v] + bias[col];
#pragma unroll
      for (int r = 0; r < RANK; ++r) acc += tv[r] * Bm[r * OUT_F + col];
      out[(size_t)row * OUT_F + col] = acc;
    }
  }
}

// ---------------------------------------------------------------------------
extern "C" void kernel_launch(void* const* d_in, const int* in_sizes, int n_in,
                              void* d_out, int out_size, void* d_ws, size_t ws_size,
                              hipStream_t stream) {
  (void)in_sizes; (void)n_in; (void)out_size; (void)ws_size;
  const float* x   = (const float*)d_in[0];
  const int*   dom = (const int*)  d_in[1];
  const float* W   = (const float*)d_in[2];
  const float* b   = (const float*)d_in[3];
  const float* Aw  = (const float*)d_in[4];
  const float* Bw  = (const float*)d_in[5];
  float* out = (float*)d_out;
  float* t   = (float*)d_ws;  // BATCH * RANK floats (256 KB) scratch

  lora_t_kernel<<<BATCH / 8, 256, 0, stream>>>(x, dom, Aw, t);

  dim3 grid(OUT_F / BN, BATCH / BM);
  lora_gemm_kernel<<<grid, 256, 0, stream>>>(x, dom, W, b, Bw, t, out);
}